// MeshEncoder_14645838479864
// MI455X (gfx1250) — compile-verified
//
#include <hip/hip_runtime.h>

#define MKER 27            // 3^3 spline kernels
#define KSPLINE (MKER * 64)  // 1728 reduction depth for 64-ch layers

typedef __attribute__((ext_vector_type(2))) float v2f;
typedef __attribute__((ext_vector_type(8))) float v8f;

// ---------------- helpers ----------------

__global__ void zero_f32(float* __restrict__ p, long n) {
  long i = (long)blockIdx.x * blockDim.x + threadIdx.x;
  long stride = (long)gridDim.x * blockDim.x;
  for (; i < n; i += stride) p[i] = 0.0f;
}

__global__ void deg_kernel(const int* __restrict__ ei, int E, float* __restrict__ deg) {
  int e = blockIdx.x * blockDim.x + threadIdx.x;
  if (e >= E) return;
  int dst = ei[E + e];
  unsafeAtomicAdd(&deg[dst], 1.0f);
}

__global__ void invdeg_kernel(float* __restrict__ deg, int N) {
  int n = blockIdx.x * blockDim.x + threadIdx.x;
  if (n >= N) return;
  deg[n] = 1.0f / fmaxf(deg[n], 1.0f);
}

// degree-1 open B-spline basis, K=3: pos in [0,2], i0 = clip(floor,0,1)
__device__ __forceinline__ void load_basis(const float* __restrict__ pseudo, long e,
                                           float& f0, float& f1, float& f2,
                                           int& i0, int& i1, int& i2) {
  float p0 = pseudo[e * 3 + 0] * 2.0f;
  float p1 = pseudo[e * 3 + 1] * 2.0f;
  float p2 = pseudo[e * 3 + 2] * 2.0f;
  float a0 = fminf(floorf(p0), 1.0f);
  float a1 = fminf(floorf(p1), 1.0f);
  float a2 = fminf(floorf(p2), 1.0f);
  i0 = (int)a0; i1 = (int)a1; i2 = (int)a2;
  f0 = p0 - a0; f1 = p1 - a1; f2 = p2 - a2;
}

__device__ __forceinline__ void corner(float f0, float f1, float f2,
                                       int i0, int i1, int i2, int c,
                                       float& w, int& kidx) {
  int o0 = c & 1, o1 = (c >> 1) & 1, o2 = (c >> 2) & 1;
  w = (o0 ? f0 : 1.0f - f0) * (o1 ? f1 : 1.0f - f1) * (o2 ? f2 : 1.0f - f2);
  kidx = (i0 + o0) + 3 * (i1 + o1) + 9 * (i2 + o2);
}

// ---------------- layer-0 scatter: C_in = 3, one wave per edge, lanes 0..23 = (corner, channel)
__global__ void scatter3_kernel(const float* __restrict__ x, const int* __restrict__ ei,
                                const float* __restrict__ pseudo,
                                float* __restrict__ acc0, int E) {
  int gw = (int)(((long)blockIdx.x * blockDim.x + threadIdx.x) >> 5);
  int lane = threadIdx.x & 31;
  if (gw >= E) return;                      // wave-uniform
  int src = ei[gw], dst = ei[E + gw];
  float f0, f1, f2; int i0, i1, i2;
  load_basis(pseudo, gw, f0, f1, f2, i0, i1, i2);
  if (lane < 24) {
    int c = lane / 3, j = lane - 3 * c;
    float w; int kk;
    corner(f0, f1, f2, i0, i1, i2, c, w, kk);
    float xv = x[(long)src * 3 + j];
    unsafeAtomicAdd(&acc0[((long)dst * MKER + kk) * 3 + j], w * xv);
  }
}

// ---------------- 64-ch scatter: one wave per edge, lane j covers channels j and j+32
__global__ void scatter64_kernel(const float* __restrict__ h, const int* __restrict__ ei,
                                 const float* __restrict__ pseudo,
                                 float* __restrict__ acc, int E) {
  int gw = (int)(((long)blockIdx.x * blockDim.x + threadIdx.x) >> 5);
  int lane = threadIdx.x & 31;
  if (gw >= E) return;                      // wave-uniform
  int src = ei[gw], dst = ei[E + gw];
  float h0 = h[(long)src * 64 + lane];
  float h1 = h[(long)src * 64 + lane + 32];
  float f0, f1, f2; int i0, i1, i2;
  load_basis(pseudo, gw, f0, f1, f2, i0, i1, i2);
#pragma unroll
  for (int c = 0; c < 8; ++c) {
    float w; int kk;
    corner(f0, f1, f2, i0, i1, i2, c, w, kk);
    float* p = &acc[((long)dst * MKER + kk) * 64];
    unsafeAtomicAdd(p + lane, w * h0);       // 32 lanes -> 128B coalesced atomic row
    unsafeAtomicAdd(p + lane + 32, w * h1);
  }
}

// ---------------- layer-0 GEMM (K=81), VALU; 4 nodes per 256-thread block, acc row in LDS
__global__ __launch_bounds__(256) void gemm3_kernel(
    const float* __restrict__ acc0, const float* __restrict__ x,
    const float* __restrict__ W0, const float* __restrict__ r0,
    const float* __restrict__ b0, const float* __restrict__ invdeg,
    float* __restrict__ out, int N) {
  __shared__ float sacc[4 * 81];
  int nb = blockIdx.x * 4;
  long lim = (long)N * 81;
  for (int idx = threadIdx.x; idx < 4 * 81; idx += 256) {
    long g = (long)nb * 81 + idx;
    sacc[idx] = (g < lim) ? acc0[g] : 0.0f;
  }
  __syncthreads();
  int ln = threadIdx.x >> 6;
  int o  = threadIdx.x & 63;
  int n  = nb + ln;
  if (n >= N) return;
  float s = invdeg[n];
  float a = 0.0f;
  for (int q = 0; q < 81; ++q) a += sacc[ln * 81 + q] * W0[q * 64 + o];
  float rv = x[(long)n * 3 + 0] * r0[0 * 64 + o]
           + x[(long)n * 3 + 1] * r0[1 * 64 + o]
           + x[(long)n * 3 + 2] * r0[2 * 64 + o];
  float v = a * s + rv + b0[o];
  out[(long)n * 64 + o] = fmaxf(v, 0.0f);
}

// ---------------- 64-ch GEMM via V_WMMA_F32_16X16X4_F32
// Each wave owns a 16-row x 64-col output tile (4 accumulators).
// Spline part: A = inv_deg-prescaled acc rows (K=1728); root part: A = h_prev (K=64).
__global__ __launch_bounds__(256) void gemm64_wmma(
    const float* __restrict__ acc, const float* __restrict__ hprev,
    const float* __restrict__ W,   const float* __restrict__ Wr,
    const float* __restrict__ bias, const float* __restrict__ invdeg,
    float* __restrict__ out, int N) {
  int wave = (int)(((long)blockIdx.x * blockDim.x + threadIdx.x) >> 5);
  int rowBase = wave * 16;
  if (rowBase >= N) return;                 // wave-uniform (EXEC all-1 inside)
  int lane = threadIdx.x & 31;
  int half = lane >> 4;                     // 0: K=2h, 1: K=2h (upper pair)
  int lcol = lane & 15;
  int r = rowBase + lcol;                   // A-matrix: lane owns row r
  float s = invdeg[r];
  const float* aRow = acc + (long)r * KSPLINE;

  v8f c0 = {}, c1 = {}, c2 = {}, c3 = {};

#pragma unroll 2
  for (int k = 0; k < KSPLINE; k += 4) {
    int ka = k + 2 * half;
    v2f a = *(const v2f*)(aRow + ka);
    a[0] *= s; a[1] *= s;                   // fold mean-by-degree into A (per-lane uniform row)
    const float* wp = W + (long)ka * 64;
    v2f bb;
    bb[0] = wp[lcol];        bb[1] = wp[64 + lcol];
    c0 = __builtin_amdgcn_wmma_f32_16x16x4_f32(false, a, false, bb, (short)0, c0, false, false);
    bb[0] = wp[16 + lcol];   bb[1] = wp[80 + lcol];
    c1 = __builtin_amdgcn_wmma_f32_16x16x4_f32(false, a, false, bb, (short)0, c1, false, false);
    bb[0] = wp[32 + lcol];   bb[1] = wp[96 + lcol];
    c2 = __builtin_amdgcn_wmma_f32_16x16x4_f32(false, a, false, bb, (short)0, c2, false, false);
    bb[0] = wp[48 + lcol];   bb[1] = wp[112 + lcol];
    c3 = __builtin_amdgcn_wmma_f32_16x16x4_f32(false, a, false, bb, (short)0, c3, false, false);
  }

  const float* hRow = hprev + (long)r * 64;
#pragma unroll 2
  for (int k = 0; k < 64; k += 4) {
    int ka = k + 2 * half;
    v2f a = *(const v2f*)(hRow + ka);       // root part: NOT degree-scaled
    const float* wp = Wr + (long)ka * 64;
    v2f bb;
    bb[0] = wp[lcol];        bb[1] = wp[64 + lcol];
    c0 = __builtin_amdgcn_wmma_f32_16x16x4_f32(false, a, false, bb, (short)0, c0, false, false);
    bb[0] = wp[16 + lcol];   bb[1] = wp[80 + lcol];
    c1 = __builtin_amdgcn_wmma_f32_16x16x4_f32(false, a, false, bb, (short)0, c1, false, false);
    bb[0] = wp[32 + lcol];   bb[1] = wp[96 + lcol];
    c2 = __builtin_amdgcn_wmma_f32_16x16x4_f32(false, a, false, bb, (short)0, c2, false, false);
    bb[0] = wp[48 + lcol];   bb[1] = wp[112 + lcol];
    c3 = __builtin_amdgcn_wmma_f32_16x16x4_f32(false, a, false, bb, (short)0, c3, false, false);
  }

  // C/D layout: VGPR v, lanes 0-15 -> row v, lanes 16-31 -> row v+8; col = lane&15
  float bb0 = bias[lcol], bb1 = bias[lcol + 16], bb2 = bias[lcol + 32], bb3 = bias[lcol + 48];
#pragma unroll
  for (int v = 0; v < 8; ++v) {
    int ro = rowBase + v + 8 * half;
    long base = (long)ro * 64 + lcol;
    out[base]      = fmaxf(c0[v] + bb0, 0.0f);
    out[base + 16] = fmaxf(c1[v] + bb1, 0.0f);
    out[base + 32] = fmaxf(c2[v] + bb2, 0.0f);
    out[base + 48] = fmaxf(c3[v] + bb3, 0.0f);
  }
}

// ---------------- driver ----------------

extern "C" void kernel_launch(void* const* d_in, const int* in_sizes, int n_in,
                              void* d_out, int out_size, void* d_ws, size_t ws_size,
                              hipStream_t stream) {
  const float* x      = (const float*)d_in[0];
  const int*   ei     = (const int*)d_in[1];
  const float* pseudo = (const float*)d_in[2];
  const float* W0 = (const float*)d_in[3];
  const float* r0 = (const float*)d_in[4];
  const float* b0 = (const float*)d_in[5];
  const float* W1 = (const float*)d_in[6];
  const float* r1 = (const float*)d_in[7];
  const float* b1 = (const float*)d_in[8];
  const float* W2 = (const float*)d_in[9];
  const float* r2 = (const float*)d_in[10];
  const float* b2 = (const float*)d_in[11];
  float* out = (float*)d_out;

  const long N = in_sizes[0] / 3;
  const long E = in_sizes[1] / 2;

  auto al = [](long v) { return (v + 63) & ~63L; };
  float* ws   = (float*)d_ws;
  float* deg  = ws;                              // N   (becomes inv_deg in place)
  float* acc0 = deg  + al(N);                    // N*81
  float* h1   = acc0 + al(N * 81);               // N*64
  float* h2   = h1   + al(N * 64);               // N*64
  float* acc  = h2   + al(N * 64);               // N*1728 (reused by layers 1 and 2)

  const int ZB = 2048;
  const long edgeThreads = E * 32;
  const int  edgeBlocks  = (int)((edgeThreads + 255) / 256);
  const long waves64     = (N + 15) / 16;
  const int  gemmBlocks  = (int)((waves64 + 7) / 8);

  // degrees + layer-0 scatter
  zero_f32<<<ZB, 256, 0, stream>>>(deg, N);
  zero_f32<<<ZB, 256, 0, stream>>>(acc0, N * 81);
  deg_kernel<<<(int)((E + 255) / 256), 256, 0, stream>>>(ei, (int)E, deg);
  scatter3_kernel<<<edgeBlocks, 256, 0, stream>>>(x, ei, pseudo, acc0, (int)E);
  invdeg_kernel<<<(int)((N + 255) / 256), 256, 0, stream>>>(deg, (int)N);
  gemm3_kernel<<<(int)((N + 3) / 4), 256, 0, stream>>>(acc0, x, W0, r0, b0, deg, h1, (int)N);

  // layer 1 (64 -> 64)
  zero_f32<<<ZB, 256, 0, stream>>>(acc, N * KSPLINE);
  scatter64_kernel<<<edgeBlocks, 256, 0, stream>>>(h1, ei, pseudo, acc, (int)E);
  gemm64_wmma<<<gemmBlocks, 256, 0, stream>>>(acc, h1, W1, r1, b1, deg, h2, (int)N);

  // layer 2 (64 -> 64) -> d_out
  zero_f32<<<ZB, 256, 0, stream>>>(acc, N * KSPLINE);
  scatter64_kernel<<<edgeBlocks, 256, 0, stream>>>(h2, ei, pseudo, acc, (int)E);
  gemm64_wmma<<<gemmBlocks, 256, 0, stream>>>(acc, h2, W2, r2, b2, deg, out, (int)N);
}